// SpatialProcessor_14714557956226
// MI455X (gfx1250) — compile-verified
//
#include <hip/hip_runtime.h>
#include <math.h>

#define N_NODES 10000
#define IN_DIM  128
#define UNITS   64
#define HEADS1  4
#define KNN     20
#define EDGES   (KNN + 1)
#define NEG_SLOPE 0.2f
#define NEG_HUGE (-1.0e30f)

// kNN tiling: 8 rows (waves) per block, 512-column LDS chunks.
#define ROWS_PER_BLOCK 8
#define CHUNK 512
#define LDS_STRIDE 20   // floats per staged column (16 data + 4 pad, keeps 16B align, 2-way conflicts)

typedef __attribute__((ext_vector_type(2))) float v2f;
typedef __attribute__((ext_vector_type(8))) float v8f;

// ---------------------------------------------------------------------------
// K0: cosine-normalize the 16-dim embeddings. One thread per node.
// ---------------------------------------------------------------------------
__global__ void normalize_kernel(const float* __restrict__ emb, float* __restrict__ nrm) {
    int i = blockIdx.x * blockDim.x + threadIdx.x;
    if (i >= N_NODES) return;
    const float4* e4 = (const float4*)emb;
    float4 v0 = e4[i * 4 + 0], v1 = e4[i * 4 + 1], v2 = e4[i * 4 + 2], v3 = e4[i * 4 + 3];
    float s = v0.x*v0.x + v0.y*v0.y + v0.z*v0.z + v0.w*v0.w
            + v1.x*v1.x + v1.y*v1.y + v1.z*v1.z + v1.w*v1.w
            + v2.x*v2.x + v2.y*v2.y + v2.z*v2.z + v2.w*v2.w
            + v3.x*v3.x + v3.y*v3.y + v3.z*v3.z + v3.w*v3.w;
    float r = 1.0f / sqrtf(s);
    float4* o4 = (float4*)nrm;
    o4[i * 4 + 0] = make_float4(v0.x*r, v0.y*r, v0.z*r, v0.w*r);
    o4[i * 4 + 1] = make_float4(v1.x*r, v1.y*r, v1.z*r, v1.w*r);
    o4[i * 4 + 2] = make_float4(v2.x*r, v2.y*r, v2.z*r, v2.w*r);
    o4[i * 4 + 3] = make_float4(v3.x*r, v3.y*r, v3.z*r, v3.w*r);
}

// ---------------------------------------------------------------------------
// K1: fused cosine-sim + top-K, LDS-tiled.
// Block = 256 threads = 8 waves; wave w owns row i = blockIdx*8 + w.
// Candidate columns are staged into LDS in 512-column chunks by the whole
// block (8x reduction of L2 traffic), then each wave's lanes scan the chunk,
// maintaining a sorted (desc) top-20 in registers (static unrolled insertion,
// no scratch). Final 32-lane merge: 20 rounds of shuffle argmax + pop.
// Tie-break: lower index wins (matches jax.lax.top_k). Self kept, as in ref.
// ---------------------------------------------------------------------------
__global__ void __launch_bounds__(ROWS_PER_BLOCK * 32)
knn_topk_kernel(const float* __restrict__ nrm, int* __restrict__ tgt) {
    __shared__ float s_nrm[CHUNK * LDS_STRIDE];

    const int tid  = threadIdx.x;
    const int lane = tid & 31;
    const int wave = tid >> 5;
    const int i    = blockIdx.x * ROWS_PER_BLOCK + wave;   // N_NODES % ROWS_PER_BLOCK == 0

    const float4* n4 = (const float4*)nrm;
    float4 q0 = n4[i * 4 + 0], q1 = n4[i * 4 + 1], q2 = n4[i * 4 + 2], q3 = n4[i * 4 + 3];

    float bv[KNN];
    int   bi[KNN];
#pragma unroll
    for (int k = 0; k < KNN; ++k) { bv[k] = NEG_HUGE; bi[k] = 0x7fffffff; }

    for (int cs = 0; cs < N_NODES; cs += CHUNK) {
        const int clen = min(CHUNK, N_NODES - cs);
        __syncthreads();                               // protect LDS reuse
        // cooperative stage: clen columns x 4 float4 each
        for (int t = tid; t < clen * 4; t += ROWS_PER_BLOCK * 32) {
            int col = t >> 2, part = t & 3;
            *(float4*)&s_nrm[col * LDS_STRIDE + part * 4] = n4[(size_t)(cs + col) * 4 + part];
        }
        __syncthreads();

        for (int jj = lane; jj < clen; jj += 32) {
            const float* p = &s_nrm[jj * LDS_STRIDE];
            float4 p0 = *(const float4*)(p + 0);
            float4 p1 = *(const float4*)(p + 4);
            float4 p2 = *(const float4*)(p + 8);
            float4 p3 = *(const float4*)(p + 12);
            float d = q0.x*p0.x + q0.y*p0.y + q0.z*p0.z + q0.w*p0.w
                    + q1.x*p1.x + q1.y*p1.y + q1.z*p1.z + q1.w*p1.w
                    + q2.x*p2.x + q2.y*p2.y + q2.z*p2.z + q2.w*p2.w
                    + q3.x*p3.x + q3.y*p3.y + q3.z*p3.z + q3.w*p3.w;
            int j = cs + jj;
            if (d > bv[KNN - 1] || (d == bv[KNN - 1] && j < bi[KNN - 1])) {
                float cv = d; int ci = j;
#pragma unroll
                for (int k = 0; k < KNN; ++k) {
                    bool sw = (cv > bv[k]) || (cv == bv[k] && ci < bi[k]);
                    float tv = sw ? bv[k] : cv;   int ti = sw ? bi[k] : ci;
                    bv[k]    = sw ? cv    : bv[k]; bi[k] = sw ? ci : bi[k];
                    cv = tv; ci = ti;
                }
            }
        }
    }

    // merge the 32 per-lane sorted lists: 20 rounds of wave argmax + pop.
    for (int r = 0; r < KNN; ++r) {
        float v = bv[0]; int id = bi[0];
#pragma unroll
        for (int off = 16; off >= 1; off >>= 1) {
            float ov = __shfl_xor(v, off, 32);
            int   oi = __shfl_xor(id, off, 32);
            if (ov > v || (ov == v && oi < id)) { v = ov; id = oi; }
        }
        if (lane == 0) tgt[i * KNN + r] = id;
        if (bv[0] == v && bi[0] == id) {     // owner lane pops its head
#pragma unroll
            for (int k = 0; k < KNN - 1; ++k) { bv[k] = bv[k + 1]; bi[k] = bi[k + 1]; }
            bv[KNN - 1] = NEG_HUGE; bi[KNN - 1] = 0x7fffffff;
        }
    }
}

// ---------------------------------------------------------------------------
// K2/K5: dense f32 GEMM C[M,N] = A[M,K] @ B[K,N] using V_WMMA_F32_16X16X4_F32.
// One wave per 16x16 C tile; K chained in steps of 4.
// A frag (16x4): lane%16 = row M, lane/16 picks K-half, VGPR0/1 = K, K+1.
// B frag (4x16): lane%16 = col N, lane/16 picks K-half, VGPR0/1 = K, K+1.
// C/D (16x16 f32): VGPR r -> row = r + (lane/16)*8, col = lane%16.
// ---------------------------------------------------------------------------
__global__ void wmma_gemm_f32_kernel(const float* __restrict__ A, const float* __restrict__ B,
                                     float* __restrict__ C, int M, int N, int K) {
    int wavesPerBlock = blockDim.x >> 5;
    int wave = blockIdx.x * wavesPerBlock + ((int)threadIdx.x >> 5);
    int ntiles = N >> 4;
    int mtiles = M >> 4;
    if (wave >= ntiles * mtiles) return;   // uniform per wave
    int tm = wave / ntiles, tn = wave % ntiles;
    int lane = threadIdx.x & 31;
    int half = lane >> 4, l16 = lane & 15;

    v8f acc = {};
    const float* arow  = A + (size_t)(tm * 16 + l16) * K;
    const float* bbase = B + tn * 16 + l16;
    for (int k = 0; k < K; k += 4) {
        int ka = k + half * 2;
        v2f a, b;
        a.x = arow[ka];
        a.y = arow[ka + 1];
        b.x = bbase[(size_t)ka * N];
        b.y = bbase[(size_t)(ka + 1) * N];
        acc = __builtin_amdgcn_wmma_f32_16x16x4_f32(
            /*neg_a=*/false, a, /*neg_b=*/false, b,
            /*c_mod=*/(short)0, acc, /*reuse_a=*/false, /*reuse_b=*/false);
    }
    float* crow = C + (size_t)(tm * 16 + half * 8) * N + tn * 16 + l16;
#pragma unroll
    for (int r = 0; r < 8; ++r) crow[(size_t)r * N] = acc[r];
}

// ---------------------------------------------------------------------------
// K3/K6: attention coefficients e_src[n,h] = <h[n,h,:], a_src[h,:]> (and tgt).
// One thread per (n,h).
// ---------------------------------------------------------------------------
template <int H, int C>
__global__ void attn_coef_kernel(const float* __restrict__ h, const float* __restrict__ a_src,
                                 const float* __restrict__ a_tgt, float* __restrict__ e_src,
                                 float* __restrict__ e_tgt) {
    int t = blockIdx.x * blockDim.x + threadIdx.x;
    if (t >= N_NODES * H) return;
    int n = t / H, hh = t % H;
    const float* hp = h + (size_t)n * (H * C) + hh * C;
    const float* as = a_src + hh * C;
    const float* at = a_tgt + hh * C;
    float s = 0.f, g = 0.f;
#pragma unroll 8
    for (int c = 0; c < C; ++c) { float v = hp[c]; s += v * as[c]; g += v * at[c]; }
    e_src[t] = s;
    e_tgt[t] = g;
}

// ---------------------------------------------------------------------------
// K4/K7: per-node GAT aggregation. Edges of node i = its 20 kNN targets plus
// the explicit self loop (segment grouping by src makes this exact).
// One workgroup per node; thread = output channel (H*C threads).
// ---------------------------------------------------------------------------
template <int H, int C, bool RELU>
__global__ void gat_aggregate_kernel(const float* __restrict__ h, const float* __restrict__ es,
                                     const float* __restrict__ et, const int* __restrict__ tgt,
                                     const float* __restrict__ bias, float* __restrict__ out) {
    int i   = blockIdx.x;
    int tid = threadIdx.x;             // 0 .. H*C-1
    __shared__ float s_alpha[EDGES * H];
    __shared__ int   s_tgt[EDGES];

    if (tid < EDGES) s_tgt[tid] = (tid < KNN) ? tgt[i * KNN + tid] : i;
    __syncthreads();

    if (tid < EDGES * H) {             // tid == e*H + hh
        int e = tid / H, hh = tid % H;
        float lg = es[i * H + hh] + et[s_tgt[e] * H + hh];
        lg = (lg > 0.f) ? lg : lg * NEG_SLOPE;   // leaky_relu
        s_alpha[tid] = lg;
    }
    __syncthreads();

    if (tid < H) {                     // segment softmax over 21 edges
        float m = NEG_HUGE;
        for (int e = 0; e < EDGES; ++e) m = fmaxf(m, s_alpha[e * H + tid]);
        float z = 0.f;
        for (int e = 0; e < EDGES; ++e) z += expf(s_alpha[e * H + tid] - m);
        float inv = 1.f / z;
        for (int e = 0; e < EDGES; ++e) s_alpha[e * H + tid] = expf(s_alpha[e * H + tid] - m) * inv;
    }
    __syncthreads();

    int hh = tid / C;
    float acc = 0.f;
#pragma unroll
    for (int e = 0; e < EDGES; ++e)
        acc += s_alpha[e * H + hh] * h[(size_t)s_tgt[e] * (H * C) + tid];
    acc += bias[tid];
    if (RELU) acc = fmaxf(acc, 0.f);
    out[(size_t)i * (H * C) + tid] = acc;
}

// ---------------------------------------------------------------------------
// Launch. Inputs: x, emb, W1, a1_src, a1_tgt, b1, W2, a2_src, a2_tgt, b2.
// ---------------------------------------------------------------------------
extern "C" void kernel_launch(void* const* d_in, const int* in_sizes, int n_in,
                              void* d_out, int out_size, void* d_ws, size_t ws_size,
                              hipStream_t stream) {
    const float* x   = (const float*)d_in[0];
    const float* emb = (const float*)d_in[1];
    const float* W1  = (const float*)d_in[2];   // [128,4,64] == [128,256] row-major
    const float* a1s = (const float*)d_in[3];
    const float* a1t = (const float*)d_in[4];
    const float* b1  = (const float*)d_in[5];
    const float* W2  = (const float*)d_in[6];   // [256,1,64] == [256,64] row-major
    const float* a2s = (const float*)d_in[7];
    const float* a2t = (const float*)d_in[8];
    const float* b2  = (const float*)d_in[9];
    float* out = (float*)d_out;

    char* ws = (char*)d_ws;
    auto alloc = [&](size_t bytes) {
        char* p = ws;
        ws += (bytes + 255) & ~(size_t)255;
        return p;
    };
    float* nrm  = (float*)alloc((size_t)N_NODES * 16 * sizeof(float));
    int*   tgt  = (int*)  alloc((size_t)N_NODES * KNN * sizeof(int));
    float* h1   = (float*)alloc((size_t)N_NODES * HEADS1 * UNITS * sizeof(float));
    float* e1s  = (float*)alloc((size_t)N_NODES * HEADS1 * sizeof(float));
    float* e1t  = (float*)alloc((size_t)N_NODES * HEADS1 * sizeof(float));
    float* agg1 = (float*)alloc((size_t)N_NODES * HEADS1 * UNITS * sizeof(float));
    float* h2   = (float*)alloc((size_t)N_NODES * UNITS * sizeof(float));
    float* e2s  = (float*)alloc((size_t)N_NODES * sizeof(float));
    float* e2t  = (float*)alloc((size_t)N_NODES * sizeof(float));

    // graph build
    normalize_kernel<<<(N_NODES + 255) / 256, 256, 0, stream>>>(emb, nrm);
    knn_topk_kernel<<<N_NODES / ROWS_PER_BLOCK, ROWS_PER_BLOCK * 32, 0, stream>>>(nrm, tgt);

    // layer 1: h1 = x @ W1  (WMMA f32), attention, aggregate + bias + relu
    {
        int tiles = (N_NODES / 16) * ((HEADS1 * UNITS) / 16);   // 625*16
        wmma_gemm_f32_kernel<<<(tiles + 7) / 8, 256, 0, stream>>>(
            x, W1, h1, N_NODES, HEADS1 * UNITS, IN_DIM);
    }
    attn_coef_kernel<HEADS1, UNITS><<<(N_NODES * HEADS1 + 255) / 256, 256, 0, stream>>>(
        h1, a1s, a1t, e1s, e1t);
    gat_aggregate_kernel<HEADS1, UNITS, true><<<N_NODES, HEADS1 * UNITS, 0, stream>>>(
        h1, e1s, e1t, tgt, b1, agg1);

    // layer 2: h2 = agg1 @ W2 (WMMA f32), attention, aggregate + bias
    {
        int tiles = (N_NODES / 16) * (UNITS / 16);              // 625*4
        wmma_gemm_f32_kernel<<<(tiles + 7) / 8, 256, 0, stream>>>(
            agg1, W2, h2, N_NODES, UNITS, HEADS1 * UNITS);
    }
    attn_coef_kernel<1, UNITS><<<(N_NODES + 255) / 256, 256, 0, stream>>>(
        h2, a2s, a2t, e2s, e2t);
    gat_aggregate_kernel<1, UNITS, false><<<N_NODES, UNITS, 0, stream>>>(
        h2, e2s, e2t, tgt, b2, out);
}